// SpikeFP32MatMul_11450382811495
// MI455X (gfx1250) — compile-verified
//
#include <hip/hip_runtime.h>

typedef __attribute__((ext_vector_type(2))) float v2f;
typedef __attribute__((ext_vector_type(8))) float v8f;

#define K_DIM 512
#define N_DIM 512
#define M_TOT 2048                    // 4 batches * 512 rows, batch folded into M
#define NA_FLOATS (M_TOT * K_DIM)     // decoded A element count (2048*512)
#define NB_FLOATS (K_DIM * N_DIM)     // decoded B element count (512*512)
#define NC_FLOATS (M_TOT * N_DIM)     // C element count (2048*512)

// ---------------------------------------------------------------------------
// Kernel 1: pulse bits -> fp32.  One thread per decoded float: reads 32 bit-
// floats (128B contiguous as 8 x float4), assembles the uint32 bit pattern.
// ---------------------------------------------------------------------------
__global__ __launch_bounds__(256) void decode_bits(const float4* __restrict__ Abits,
                                                   const float4* __restrict__ Bbits,
                                                   float* __restrict__ Af,
                                                   float* __restrict__ Bf) {
  int t = blockIdx.x * blockDim.x + threadIdx.x;
  const float4* src;
  float* dst;
  int idx;
  if (t < NA_FLOATS) {
    src = Abits; dst = Af; idx = t;
  } else {
    idx = t - NA_FLOATS;
    if (idx >= NB_FLOATS) return;
    src = Bbits; dst = Bf;
  }
  const float4* p = src + (size_t)idx * 8;   // 32 bits = 8 float4
  unsigned u = 0u;
#pragma unroll
  for (int q = 0; q < 8; ++q) {
    float4 v = p[q];
    u |= (v.x != 0.0f ? 1u : 0u) << (4 * q + 0);
    u |= (v.y != 0.0f ? 1u : 0u) << (4 * q + 1);
    u |= (v.z != 0.0f ? 1u : 0u) << (4 * q + 2);
    u |= (v.w != 0.0f ? 1u : 0u) << (4 * q + 3);
  }
  dst[idx] = __uint_as_float(u);
}

// ---------------------------------------------------------------------------
// Kernel 2: fp32 GEMM via V_WMMA_F32_16X16X4_F32 (precision-preserving path).
// One wave32 per 16x16 tile of C.  Fragment layouts per CDNA5 ISA 7.12.2:
//   A 16x4:  lanes 0-15 -> M=lane, {K=0 (v0), K=1 (v1)};
//            lanes 16-31 -> M=lane-16, {K=2 (v0), K=3 (v1)}
//   B 4x16:  mirrored (N across lanes, K pair selected by lane half)
//   C 16x16: VGPR r -> M = r + 8*half, N = lane&15
// Grid exactly covers all tiles -> EXEC is all-ones for every WMMA.
// ---------------------------------------------------------------------------
__global__ __launch_bounds__(128) void wmma_gemm(const float* __restrict__ Af,
                                                 const float* __restrict__ Bf,
                                                 float* __restrict__ Cf) {
  int wave = blockIdx.x * (blockDim.x >> 5) + (threadIdx.x >> 5);
  int lane = threadIdx.x & 31;
  int mt = wave >> 5;   // 0..127  (N_DIM/16 = 32 n-tiles per m-tile row)
  int nt = wave & 31;   // 0..31
  int m0 = mt * 16;
  int n0 = nt * 16;
  int half = lane >> 4;     // which K pair this lane holds
  int l = lane & 15;

  v8f c = {};
  const float* arow = Af + (size_t)(m0 + l) * K_DIM;
  const float* bcol = Bf + n0 + l;

#pragma unroll 4
  for (int k0 = 0; k0 < K_DIM; k0 += 4) {
    int kk = k0 + 2 * half;
    v2f a, b;
    a.x = arow[kk];
    a.y = arow[kk + 1];
    b.x = bcol[(size_t)kk * N_DIM];
    b.y = bcol[(size_t)(kk + 1) * N_DIM];
    // 8 args: (neg_a, A, neg_b, B, c_mod, C, reuse_a, reuse_b)
    c = __builtin_amdgcn_wmma_f32_16x16x4_f32(false, a, false, b,
                                              (short)0, c, false, false);
  }

  float* crow = Cf + (size_t)(m0 + 8 * half) * N_DIM + n0 + l;
#pragma unroll
  for (int r = 0; r < 8; ++r)
    crow[(size_t)r * N_DIM] = c[r];
}

// ---------------------------------------------------------------------------
// Kernel 3: fp32 -> pulse bits.  One thread per C value, 128B contiguous
// stores (8 x float4 of {0.0, 1.0}).
// ---------------------------------------------------------------------------
__global__ __launch_bounds__(256) void encode_bits(const float* __restrict__ Cf,
                                                   float4* __restrict__ out) {
  int t = blockIdx.x * blockDim.x + threadIdx.x;
  if (t >= NC_FLOATS) return;
  unsigned u = __float_as_uint(Cf[t]);
  float4* p = out + (size_t)t * 8;
#pragma unroll
  for (int q = 0; q < 8; ++q) {
    float4 v;
    v.x = ((u >> (4 * q + 0)) & 1u) ? 1.0f : 0.0f;
    v.y = ((u >> (4 * q + 1)) & 1u) ? 1.0f : 0.0f;
    v.z = ((u >> (4 * q + 2)) & 1u) ? 1.0f : 0.0f;
    v.w = ((u >> (4 * q + 3)) & 1u) ? 1.0f : 0.0f;
    p[q] = v;
  }
}

extern "C" void kernel_launch(void* const* d_in, const int* in_sizes, int n_in,
                              void* d_out, int out_size, void* d_ws, size_t ws_size,
                              hipStream_t stream) {
  const float4* Abits = (const float4*)d_in[0];   // [4,512,512,32] pulses
  const float4* Bbits = (const float4*)d_in[1];   // [512,512,32] pulses

  char* ws = (char*)d_ws;
  float* Af = (float*)(ws);                                        // 4 MB
  float* Bf = (float*)(ws + (size_t)NA_FLOATS * 4);                // 1 MB
  float* Cf = (float*)(ws + (size_t)(NA_FLOATS + NB_FLOATS) * 4);  // 4 MB

  int nDec = NA_FLOATS + NB_FLOATS;                 // 1,310,720
  decode_bits<<<(nDec + 255) / 256, 256, 0, stream>>>(Abits, Bbits, Af, Bf);

  // 4096 tiles (128 m-tiles x 32 n-tiles), 4 waves per 128-thread block
  wmma_gemm<<<1024, 128, 0, stream>>>(Af, Bf, Cf);

  encode_bits<<<(NC_FLOATS + 255) / 256, 256, 0, stream>>>(Cf, (float4*)d_out);
}